// S2Network_24653112279617
// MI455X (gfx1250) — compile-verified
//
#include <hip/hip_runtime.h>
#include <math.h>

// ---------------------------------------------------------------------------
// S2-equivariant network forward for MI455X (gfx1250).
//   * Dominant cost (S2 grid transform, ~10 GFLOP) -> fused f32 WMMA kernel
//     (V_WMMA_F32_16X16X4_F32), one wave32 per sample, Y streamed via LDS.
//     Fusion avoids a 210 MB intermediate that would not fit the 192 MB L2.
//   * SiLU uses v_rcp_f32 (not IEEE divide); the per-chunk quadrature weight
//     is uniform (16 | 80) and folded into the SiLU output for free.
//   * Y stream is software-pipelined; no divergent branches inside the WMMA
//     loop (EXEC stays all-ones, as WMMA requires).
//   * Sparse CG tensor-square + linear (~2 GFLOP) -> VALU kernel.
//   * All constants (CG, Gauss-Legendre grid, real SH basis) built on device
//     each launch: deterministic and graph-capture safe.
// ---------------------------------------------------------------------------

#define PI_D 3.14159265358979323846
#define NRES 80            // 20*(LMAX+1)
#define NSAMP 512
#define CG_TOTAL 2351      // sum over (l1<=l2,lo) of (2l1+1)(2l2+1)(2lo+1)
#define LPAD 20            // LDS row stride (floats): 16B-aligned, conflict-free
#define NCHUNK 400         // 80*80/16 grid chunks of 16 points

__constant__ int d_off[5]    = {0, 1, 4, 9, 16};       // OFF_ACT
__constant__ int d_offin[5]  = {0, 16, 64, 144, 256};  // OFF_IN
__constant__ int d_catmul[4] = {560, 1192, 1448, 1312};

typedef float v2f __attribute__((ext_vector_type(2)));
typedef float v8f __attribute__((ext_vector_type(8)));

// ------------------------------ math helpers -------------------------------
__device__ double d_fact(int n) { double r = 1.0; for (int i = 2; i <= n; i++) r *= i; return r; }

__device__ double d_cgc(int j1, int m1, int j2, int m2, int j3, int m3) {
    if (m1 + m2 != m3) return 0.0;
    double pre = sqrt((2 * j3 + 1) * d_fact(j3 + j1 - j2) * d_fact(j3 - j1 + j2) *
                      d_fact(j1 + j2 - j3) / d_fact(j1 + j2 + j3 + 1));
    pre *= sqrt(d_fact(j3 + m3) * d_fact(j3 - m3) * d_fact(j1 - m1) * d_fact(j1 + m1) *
                d_fact(j2 - m2) * d_fact(j2 + m2));
    int kmin = max(0, max(-(j3 - j2 + m1), -(j3 - j1 - m2)));
    int kmax = min(j1 + j2 - j3, min(j1 - m1, j2 + m2));
    double s = 0.0;
    for (int k = kmin; k <= kmax; k++) {
        double sign = (k & 1) ? -1.0 : 1.0;
        s += sign / (d_fact(k) * d_fact(j1 + j2 - j3 - k) * d_fact(j1 - m1 - k) *
                     d_fact(j2 + m2 - k) * d_fact(j3 - j2 + m1 + k) * d_fact(j3 - j1 - m2 + k));
    }
    return pre * s;
}

struct cplx { double re, im; };
__device__ inline cplx cmul(cplx a, cplx b) { return {a.re * b.re - a.im * b.im, a.re * b.im + a.im * b.re}; }
__device__ inline cplx conjc(cplx a) { return {a.re, -a.im}; }

// nonzero entries of row `a` of the real->complex change-of-basis _q(l)
__device__ int q_row(int l, int a, int cols[2], cplx vals[2]) {
    const double s = 0.70710678118654752440;
    cplx ph;  // (-i)^l
    switch (l & 3) { case 0: ph = {1, 0}; break; case 1: ph = {0, -1}; break;
                     case 2: ph = {-1, 0}; break; default: ph = {0, 1}; }
    if (a == l) { cols[0] = l; vals[0] = ph; return 1; }
    if (a < l) {
        int m = l - a; double sgn = (m & 1) ? -1.0 : 1.0;
        cols[0] = a;         vals[0] = cmul(ph, (cplx){0.0, s});
        cols[1] = 2 * l - a; vals[1] = cmul(ph, (cplx){0.0, -sgn * s});
        return 2;
    } else {
        int m = a - l; double sgn = (m & 1) ? -1.0 : 1.0;
        cols[0] = 2 * l - a; vals[0] = cmul(ph, (cplx){s, 0.0});
        cols[1] = a;         vals[1] = cmul(ph, (cplx){sgn * s, 0.0});
        return 2;
    }
}

__device__ float cg_real_elem(int l1, int l2, int l3, int a, int b, int c) {
    int c1[2], c2[2], c3[2]; cplx v1[2], v2[2], v3[2];
    int n1 = q_row(l1, a, c1, v1), n2 = q_row(l2, b, c2, v2), n3 = q_row(l3, c, c3, v3);
    cplx z = {0, 0};
    for (int p = 0; p < n1; p++)
        for (int q = 0; q < n2; q++)
            for (int r = 0; r < n3; r++) {
                int m1 = c1[p] - l1, m2 = c2[q] - l2, m3 = c3[r] - l3;
                if (m1 + m2 != m3) continue;
                double C = d_cgc(l1, m1, l2, m2, l3, m3);
                if (C == 0.0) continue;
                cplx t = cmul(cmul(v1[p], v2[q]), conjc(v3[r]));
                z.re += t.re * C; z.im += t.im * C;
            }
    return (float)z.re;
}

__device__ int cg_offset(int L1, int L2, int LO) {
    int off = 0;
    for (int l1 = 0; l1 < 4; l1++)
        for (int l2 = l1; l2 < 4; l2++) {
            int lmax = min(3, l1 + l2);
            for (int lo = l2 - l1; lo <= lmax; lo++) {
                if (l1 == L1 && l2 == L2 && lo == LO) return off;
                off += (2 * l1 + 1) * (2 * l2 + 1) * (2 * lo + 1);
            }
        }
    return 0;
}

__device__ double assoc_P(int l, int m, double x) {
    double pmm = 1.0;
    double somx2 = sqrt(fmax(1.0 - x * x, 0.0));
    for (int i = 1; i <= m; i++) pmm *= (2 * i - 1) * somx2;
    if (l == m) return pmm;
    double pmmp1 = x * (2 * m + 1) * pmm;
    for (int ll = m + 2; ll <= l; ll++) {
        double t = ((2 * ll - 1) * x * pmmp1 - (ll + m - 1) * pmm) / (ll - m);
        pmm = pmmp1; pmmp1 = t;
    }
    return pmmp1;
}

// ------------------------------ init kernels -------------------------------
__global__ __launch_bounds__(256) void init_cg(float* CG) {
    int tid = blockIdx.x * blockDim.x + threadIdx.x;
    if (tid >= CG_TOTAL) return;
    int off = 0;
    for (int l1 = 0; l1 < 4; l1++)
        for (int l2 = l1; l2 < 4; l2++) {
            int lmax = min(3, l1 + l2);
            for (int lo = l2 - l1; lo <= lmax; lo++) {
                int dk = 2 * lo + 1, dj = 2 * l2 + 1;
                int sz = (2 * l1 + 1) * dj * dk;
                if (tid < off + sz) {
                    int loc = tid - off;
                    int k = loc % dk, j = (loc / dk) % dj, i = loc / (dk * dj);
                    CG[tid] = cg_real_elem(l1, l2, lo, i, j, k);
                    return;
                }
                off += sz;
            }
        }
}

__global__ __launch_bounds__(128) void init_nodes(double* nodes, float* qwb) {
    int b = threadIdx.x;
    if (b >= NRES) return;
    const int NN = NRES;
    double x = cos(PI_D * (b + 0.75) / (NN + 0.5));
    double pp = 1.0;
    for (int it = 0; it < 60; it++) {
        double p0 = 1.0, p1 = x;
        for (int k = 2; k <= NN; k++) { double p2 = ((2 * k - 1) * x * p1 - (k - 1) * p0) / k; p0 = p1; p1 = p2; }
        pp = NN * (x * p1 - p0) / (x * x - 1.0);
        double dx = p1 / pp;
        x -= dx;
        if (fabs(dx) < 1e-15) break;
    }
    { // final derivative at converged x
        double p0 = 1.0, p1 = x;
        for (int k = 2; k <= NN; k++) { double p2 = ((2 * k - 1) * x * p1 - (k - 1) * p0) / k; p0 = p1; p1 = p2; }
        pp = NN * (x * p1 - p0) / (x * x - 1.0);
    }
    double w = 2.0 / ((1.0 - x * x) * pp * pp);
    nodes[b] = x;
    qwb[b] = (float)(w * (2.0 * PI_D / NRES));
}

__global__ __launch_bounds__(256) void init_y(const double* nodes, float* Ygrid) {
    int idx = blockIdx.x * blockDim.x + threadIdx.x;
    if (idx >= NRES * NRES * 16) return;
    int d = idx & 15, g = idx >> 4, a = g % NRES, b = g / NRES;
    int l = (d < 1) ? 0 : (d < 4) ? 1 : (d < 9) ? 2 : 3;
    int m = d - d_off[l] - l;
    int am = m < 0 ? -m : m;
    double x = nodes[b];
    double P = assoc_P(l, am, x);
    double nlm = sqrt((2 * l + 1) / (4.0 * PI_D) * d_fact(l - am) / d_fact(l + am));
    double alpha = 2.0 * PI_D * a / NRES;
    double v;
    if (m == 0)      v = nlm * P;
    else if (m > 0)  v = sqrt(2.0) * ((m  & 1) ? -1.0 : 1.0) * nlm * P * cos(m * alpha);
    else             v = sqrt(2.0) * ((am & 1) ? -1.0 : 1.0) * nlm * P * sin(am * alpha);
    Ygrid[idx] = (float)v;
}

// ------------------------- repack x -> coeff layout ------------------------
// x[n][ OFF_IN[l] + u*(2l+1) + i ]  ->  coeff[n][u][ OFF_ACT[l]+i ]
__global__ __launch_bounds__(256) void repack_kernel(const float* __restrict__ x, float* __restrict__ C) {
    int e = blockIdx.x * blockDim.x + threadIdx.x;
    if (e >= NSAMP * 256) return;
    int n = e >> 8, r = e & 255, u = r >> 4, d = r & 15;
    int l = (d < 1) ? 0 : (d < 4) ? 1 : (d < 9) ? 2 : 3;
    int i = d - d_off[l];
    C[e] = x[n * 256 + d_offin[l] + u * (2 * l + 1) + i];
}

// ------------------ fused tensor-square + linear (VALU) --------------------
// acc[k] = sum_u W[u]*X[u][off_lo+k] + sum_pairs sum_{uv} W[col] * z_uv[k]
__device__ __forceinline__ void tp_lin(const float* __restrict__ X, const float* __restrict__ Wr,
                                       const float* __restrict__ CG, int lo, float* acc) {
    const int dim_o = 2 * lo + 1;
    for (int k = 0; k < dim_o; k++) acc[k] = 0.f;
    // self channels
    for (int u = 0; u < 16; u++) {
        float w = Wr[u];
        const float* xr = X + u * 16 + d_off[lo];
        for (int k = 0; k < dim_o; k++) acc[k] += w * xr[k];
    }
    int col = 16;
    for (int l1 = 0; l1 < 4; l1++)
        for (int l2 = l1; l2 < 4; l2++) {
            if (lo < l2 - l1 || lo > l1 + l2) continue;
            const int dd1 = 2 * l1 + 1, dd2 = 2 * l2 + 1;
            const float* cg = CG + cg_offset(l1, l2, lo);
            const int o1 = d_off[l1], o2 = d_off[l2];
            const bool sym = (l1 == l2);
            for (int u = 0; u < 16; u++) {
                const float* xu = X + u * 16 + o1;
                for (int v = sym ? u : 0; v < 16; v++) {
                    const float* xv = X + v * 16 + o2;
                    float z[7];
                    for (int k = 0; k < dim_o; k++) z[k] = 0.f;
                    const float* ce = cg;
                    for (int i = 0; i < dd1; i++) {
                        float xi = xu[i];
                        for (int j = 0; j < dd2; j++) {
                            float p = xi * xv[j];
                            for (int k = 0; k < dim_o; k++) z[k] += ce[k] * p;
                            ce += dim_o;
                        }
                    }
                    float w = Wr[col++];
                    for (int k = 0; k < dim_o; k++) acc[k] += w * z[k];
                }
            }
        }
}

__global__ __launch_bounds__(256) void layer_kernel(const float* __restrict__ Xin,
                                                    const float* W0, const float* W1,
                                                    const float* W2, const float* W3,
                                                    const float* __restrict__ CG,
                                                    float* __restrict__ Yout) {
    __shared__ float Xs[16][256];
    int t = threadIdx.x, s = t >> 4, o = t & 15;
    int n = blockIdx.x * 16 + s;
    const float* src = Xin + (n * 16 + o) * 16;
    for (int d = 0; d < 16; d++) Xs[s][o * 16 + d] = src[d];
    __syncthreads();
    const float* Ws[4] = {W0, W1, W2, W3};
    const float* X = &Xs[s][0];
    for (int lo = 0; lo < 4; lo++) {
        int cm = d_catmul[lo];
        float acc[7];
        tp_lin(X, Ws[lo] + o * cm, CG, lo, acc);
        float sc = rsqrtf((float)cm);
        int dim_o = 2 * lo + 1;
        for (int k = 0; k < dim_o; k++)
            Yout[(n * 16 + o) * 16 + d_off[lo] + k] = acc[k] * sc;
    }
}

// --------------- fused S2 activation: WMMA f32 16x16x4 path ----------------
// f = coeff @ Y^T ; f = silu(f) ; coeff' = f @ (Y*qw)   -- fused per sample.
// One wave32 per sample; 8 waves per block; Y streamed in 16-row chunks with
// a one-chunk software pipeline. The chunk's quadrature weight is uniform
// (16 divides 80) and is folded into the SiLU output, so GEMM2's A-fragment
// is a plain LDS read. No divergent branches inside the loop.
__global__ __launch_bounds__(256) void s2act_kernel(const float* __restrict__ Cin,
                                                    const float* __restrict__ Y,
                                                    const float* __restrict__ QWB,
                                                    float* __restrict__ Cout) {
    __shared__ float qw_s[NRES];
    __shared__ float Xs[8][256];
    __shared__ float Ych[8][16 * LPAD];
    __shared__ float Fch[8][16 * LPAD];
    int t = threadIdx.x;
    if (t < NRES) qw_s[t] = QWB[t];
    int wv = t >> 5, lane = t & 31;
    int n = blockIdx.x * 8 + wv;
    const float4* cn = (const float4*)(Cin + n * 256);
    ((float4*)&Xs[wv][0])[lane * 2 + 0] = cn[lane * 2 + 0];
    ((float4*)&Xs[wv][0])[lane * 2 + 1] = cn[lane * 2 + 1];
    __syncthreads();

    const int m = lane & 15, hi = lane >> 4;
    // A fragments for GEMM1: A = coeff (M=channel, K=d), fixed for the loop
    v2f aC[4];
    for (int j = 0; j < 4; j++) {
        aC[j].x = Xs[wv][m * 16 + 4 * j + 2 * hi];
        aC[j].y = Xs[wv][m * 16 + 4 * j + 2 * hi + 1];
    }
    v8f accT = {0.f, 0.f, 0.f, 0.f, 0.f, 0.f, 0.f, 0.f};  // out^T accumulator
    float* Yl = &Ych[wv][0];
    float* Fl = &Fch[wv][0];
    const int r2 = lane >> 1, h8 = (lane & 1) * 8;  // each lane owns half a Y row

    // pipeline prologue: fetch chunk 0 into registers
    float4 y0 = ((const float4*)(Y + r2 * 16 + h8))[0];
    float4 y1 = ((const float4*)(Y + r2 * 16 + h8))[1];

    for (int gc = 0; gc < NCHUNK; gc++) {
        int g0 = gc * 16;
        // all 16 rows of this chunk share one Gauss-Legendre row -> one weight
        float wq = qw_s[gc / 5];
        // stage current chunk to LDS (16B-aligned rows)
        ((float4*)(Yl + r2 * LPAD + h8))[0] = y0;
        ((float4*)(Yl + r2 * LPAD + h8))[1] = y1;
        // issue next chunk's loads now (clamped address, branch-free); they
        // overlap the WMMAs below
        {
            int gn = (gc + 1 < NCHUNK) ? gc + 1 : NCHUNK - 1;
            const float4* np = (const float4*)(Y + (gn * 16 + r2) * 16 + h8);
            __builtin_prefetch(Y + ((gn + 1) * 16 + r2) * 16 + h8, 0, 3);
            y0 = np[0];
            y1 = np[1];
        }
        asm volatile("s_wait_dscnt 0x0" ::: "memory");

        // GEMM1: F[c][g] = sum_d coeff[c][d] * Y[g][d]
        v8f f = {0.f, 0.f, 0.f, 0.f, 0.f, 0.f, 0.f, 0.f};
        for (int j = 0; j < 4; j++) {
            v2f b;
            b.x = Yl[m * LPAD + 4 * j + 2 * hi];
            b.y = Yl[m * LPAD + 4 * j + 2 * hi + 1];
            f = __builtin_amdgcn_wmma_f32_16x16x4_f32(false, aC[j], false, b, (short)0, f, false, false);
        }
        // SiLU via v_rcp_f32, scaled by the chunk quadrature weight; store F
        // tile for the D-fragment -> B-fragment layout change.
        for (int e = 0; e < 8; e++) {
            float xv = f[e];
            xv = xv * wq * __builtin_amdgcn_rcpf(1.f + __expf(-xv));
            Fl[(e + 8 * hi) * LPAD + m] = xv;
        }
        asm volatile("s_wait_dscnt 0x0" ::: "memory");

        // GEMM2 (transposed): out^T[d][c] += sum_g Y[g][d] * (qw*silu(F))[c][g]
        for (int tt = 0; tt < 4; tt++) {
            int r0 = 4 * tt + 2 * hi;
            v2f a, b;
            a.x = Yl[r0 * LPAD + m];
            a.y = Yl[(r0 + 1) * LPAD + m];
            b.x = Fl[m * LPAD + 4 * tt + 2 * hi];
            b.y = Fl[m * LPAD + 4 * tt + 2 * hi + 1];
            accT = __builtin_amdgcn_wmma_f32_16x16x4_f32(false, a, false, b, (short)0, accT, false, false);
        }
        (void)g0;
    }
    // out^T[d = e+8*hi][c = m] -> coeff[n][c][d]
    float* dst = Cout + n * 256;
    for (int e = 0; e < 8; e++) dst[m * 16 + e + 8 * hi] = accT[e];
}

// --------------------------------- tail ------------------------------------
__global__ __launch_bounds__(256) void tail_kernel(const float* __restrict__ Xin,
                                                   const float* __restrict__ wt0,
                                                   const float* __restrict__ wt1,
                                                   const float* __restrict__ CG,
                                                   float* __restrict__ out) {
    __shared__ float Xs[16][256];
    int t = threadIdx.x, s = t >> 4, r = t & 15;
    int n = blockIdx.x * 16 + s;
    const float* src = Xin + (n * 16 + r) * 16;
    for (int d = 0; d < 16; d++) Xs[s][r * 16 + d] = src[d];
    __syncthreads();
    const float* X = &Xs[s][0];
    if (r == 0) {
        float acc[7];
        tp_lin(X, wt0, CG, 0, acc);
        out[n * 4 + 0] = acc[0] * rsqrtf(560.f);
    } else if (r == 1) {
        float acc[7];
        tp_lin(X, wt1, CG, 1, acc);
        float sc = rsqrtf(1192.f);
        for (int k = 0; k < 3; k++) out[n * 4 + 1 + k] = acc[k] * sc;
    }
}

// ------------------------------- launcher ----------------------------------
extern "C" void kernel_launch(void* const* d_in, const int* in_sizes, int n_in,
                              void* d_out, int out_size, void* d_ws, size_t ws_size,
                              hipStream_t stream) {
    (void)in_sizes; (void)n_in; (void)out_size; (void)ws_size;
    const float* x = (const float*)d_in[0];
    const float* W[12];
    for (int i = 0; i < 12; i++) W[i] = (const float*)d_in[1 + i];
    const float* wt0 = (const float*)d_in[13];
    const float* wt1 = (const float*)d_in[14];
    float* out = (float*)d_out;
    float* ws = (float*)d_ws;

    // workspace layout (float offsets)
    float*  CG    = ws + 0;                       // 2351 floats
    float*  QWB   = ws + 2368;                    // 80 floats
    double* nodes = (double*)(ws + 2448);         // 80 doubles (160 floats)
    float*  Yg    = ws + 2624;                    // 80*80*16 = 102400 floats
    float*  bufA  = ws + 105472;                  // 512*16*16 floats
    float*  bufB  = ws + 236544;                  // 512*16*16 floats

    init_cg<<<(CG_TOTAL + 255) / 256, 256, 0, stream>>>(CG);
    init_nodes<<<1, 128, 0, stream>>>(nodes, QWB);
    init_y<<<(NRES * NRES * 16 + 255) / 256, 256, 0, stream>>>(nodes, Yg);
    repack_kernel<<<(NSAMP * 256 + 255) / 256, 256, 0, stream>>>(x, bufA);

    for (int li = 0; li < 3; li++) {
        layer_kernel<<<NSAMP / 16, 256, 0, stream>>>(bufA, W[4 * li + 0], W[4 * li + 1],
                                                     W[4 * li + 2], W[4 * li + 3], CG, bufB);
        s2act_kernel<<<NSAMP / 8, 256, 0, stream>>>(bufB, Yg, QWB, bufA);
    }
    tail_kernel<<<NSAMP / 16, 256, 0, stream>>>(bufA, wt0, wt1, CG, out);
}